// Sage_76046690943450
// MI455X (gfx1250) — compile-verified
//
#include <hip/hip_runtime.h>

#define NN 100000
#define NE 1600000
#define D  64

typedef float v2f __attribute__((ext_vector_type(2)));
typedef float v8f __attribute__((ext_vector_type(8)));

// ---------------- degree: deg[dst] += 1 ----------------
__global__ void deg_kernel(const int* __restrict__ dst, float* __restrict__ deg, int nE) {
    int e = blockIdx.x * blockDim.x + threadIdx.x;
    if (e < nE) atomicAdd(&deg[dst[e]], 1.0f);
}

// ---------------- gather: agg[dst,:] += h[src,:] ----------------
// 16 lanes per edge, 4 channels per lane (float4 gather, 4 scalar f32 atomics)
__global__ void gather_kernel(const float* __restrict__ h, const int* __restrict__ src,
                              const int* __restrict__ dst, float* __restrict__ agg, int nE) {
    long idx = (long)blockIdx.x * blockDim.x + threadIdx.x;
    int e  = (int)(idx >> 4);
    int c4 = ((int)idx & 15) * 4;
    if (e >= nE) return;
    int s = src[e];
    int d = dst[e];
    const float4 v = *(const float4*)(h + (size_t)s * D + c4);
    float* base = agg + (size_t)d * D + c4;
    atomicAdd(base + 0, v.x);
    atomicAdd(base + 1, v.y);
    atomicAdd(base + 2, v.z);
    atomicAdd(base + 3, v.w);
}

// ---------------- fused SAGE layer ----------------
// OUT[i,:] = relu?( H[i,:] @ Wself + (AGG[i,:]/max(deg,1)) @ Wneigh + b )
// One wave -> one 16x16 output tile, K=64 via 16x (V_WMMA_F32_16X16X4_F32) per matrix.
// Block = 256 threads = 8 waves = 2 M-tiles x 4 N-tiles = 32 rows x 64 cols.
// Safe for OUT==H in-place: all H reads happen before __syncthreads(), writes after.
__global__ __launch_bounds__(256) void sage_gemm_kernel(
    const float* __restrict__ H, const float* __restrict__ AGG,
    const float* __restrict__ deg,
    const float* __restrict__ Wself, const float* __restrict__ Wneigh,
    const float* __restrict__ bias, float* __restrict__ OUT, int do_relu) {

    const int lane   = threadIdx.x & 31;
    const int wave   = threadIdx.x >> 5;
    const int m_tile = wave >> 2;   // 0..1
    const int n_tile = wave & 3;    // 0..3
    const int lm     = lane & 15;
    const int hi     = lane >> 4;   // 0/1: K-half selector for A/B fragments
    const int row    = blockIdx.x * 32 + m_tile * 16 + lm;   // A-matrix row for this lane
    const int col    = n_tile * 16 + lm;                     // B/C column for this lane

    // Preload ALL self A-fragments (entire K) so in-place OUT==H is race-free.
    // ISA 16x4 f32 A layout: lanes 0-15 hold K={0,1}, lanes 16-31 hold K={2,3}
    // => per k-step kk: lane holds contiguous pair H[row][kk+2*hi .. +1].
    v2f aH[16];
#pragma unroll
    for (int s = 0; s < 16; ++s)
        aH[s] = *(const v2f*)(H + (size_t)row * D + 4 * s + 2 * hi);

    const float invd = 1.0f / fmaxf(deg[row], 1.0f);

    __syncthreads();  // all H reads by this block complete before any OUT writes

    // C/D 16x16 f32 layout: VGPR v, lanes 0-15 -> (M=v, N=lm); lanes 16-31 -> (M=v+8, N=lm)
    v8f c;
    const float bv = bias[col];
#pragma unroll
    for (int v = 0; v < 8; ++v) c[v] = bv;

#pragma unroll
    for (int s = 0; s < 16; ++s) {
        const int k = 4 * s + 2 * hi;
        // B 4x16 layout mirrors A: lanes 0-15 hold rows K={0,1}, lanes 16-31 rows K={2,3}
        v2f bS;
        bS.x = Wself[(size_t)k * D + col];
        bS.y = Wself[(size_t)(k + 1) * D + col];
        c = __builtin_amdgcn_wmma_f32_16x16x4_f32(false, aH[s], false, bS,
                                                  (short)0, c, false, false);
        v2f aA = *(const v2f*)(AGG + (size_t)row * D + 4 * s + 2 * hi);
        aA.x *= invd;
        aA.y *= invd;
        v2f bN;
        bN.x = Wneigh[(size_t)k * D + col];
        bN.y = Wneigh[(size_t)(k + 1) * D + col];
        c = __builtin_amdgcn_wmma_f32_16x16x4_f32(false, aA, false, bN,
                                                  (short)0, c, false, false);
    }

    const int out_row_base = blockIdx.x * 32 + m_tile * 16 + 8 * hi;
#pragma unroll
    for (int v = 0; v < 8; ++v) {
        float val = c[v];
        if (do_relu) val = fmaxf(val, 0.0f);
        OUT[(size_t)(out_row_base + v) * D + col] = val;
    }
}

extern "C" void kernel_launch(void* const* d_in, const int* in_sizes, int n_in,
                              void* d_out, int out_size, void* d_ws, size_t ws_size,
                              hipStream_t stream) {
    const float* feat = (const float*)d_in[0];
    const float* Ws0  = (const float*)d_in[1];
    const float* Wn0  = (const float*)d_in[2];
    const float* b0   = (const float*)d_in[3];
    const float* Ws1  = (const float*)d_in[4];
    const float* Wn1  = (const float*)d_in[5];
    const float* b1   = (const float*)d_in[6];
    const float* Ws2  = (const float*)d_in[7];
    const float* Wn2  = (const float*)d_in[8];
    const float* b2   = (const float*)d_in[9];
    const int*   src  = (const int*)d_in[10];
    const int*   dst  = (const int*)d_in[11];
    float*       out  = (float*)d_out;

    // workspace: deg [NN] | agg [NN*D] | hA [NN*D]  (~51.6 MB)
    float* deg = (float*)d_ws;
    float* agg = deg + NN;
    float* hA  = agg + (size_t)NN * D;

    const int edgeBlocks   = (NE + 255) / 256;        // deg kernel
    const int gatherBlocks = (int)((size_t)NE * 16 / 256); // 100000, exact
    const int gemmBlocks   = NN / 32;                 // 3125, exact

    hipMemsetAsync(deg, 0, (size_t)NN * sizeof(float), stream);
    deg_kernel<<<edgeBlocks, 256, 0, stream>>>(dst, deg, NE);

    // ---- layer 0: feat -> hA (relu) ----
    hipMemsetAsync(agg, 0, (size_t)NN * D * sizeof(float), stream);
    gather_kernel<<<gatherBlocks, 256, 0, stream>>>(feat, src, dst, agg, NE);
    sage_gemm_kernel<<<gemmBlocks, 256, 0, stream>>>(feat, agg, deg, Ws0, Wn0, b0, hA, 1);

    // ---- layer 1: hA -> out (relu) ----
    hipMemsetAsync(agg, 0, (size_t)NN * D * sizeof(float), stream);
    gather_kernel<<<gatherBlocks, 256, 0, stream>>>(hA, src, dst, agg, NE);
    sage_gemm_kernel<<<gemmBlocks, 256, 0, stream>>>(hA, agg, deg, Ws1, Wn1, b1, out, 1);

    // ---- layer 2: out -> out (in-place, no relu) ----
    hipMemsetAsync(agg, 0, (size_t)NN * D * sizeof(float), stream);
    gather_kernel<<<gatherBlocks, 256, 0, stream>>>(out, src, dst, agg, NE);
    sage_gemm_kernel<<<gemmBlocks, 256, 0, stream>>>(out, agg, deg, Ws2, Wn2, b2, out, 0);
}